// DMPNet_76012331205204
// MI455X (gfx1250) — compile-verified
//
#include <hip/hip_runtime.h>
#include <math.h>

// ---- problem constants ----
#define BATCH   4096
#define D_IN    64
#define D_HID   256
#define DOF     7
#define NB      30          // RBF basis count
#define OUTC    217         // NB*DOF + DOF
#define NE      (BATCH*DOF) // 28672 elements
#define NCHK    51          // T+1 checkpoints
#define STEPS   500
#define A_Z     15.0f
#define A_X     1.0f
#define TAU     1.0f
#define DT      (1.0f/500.0f)

typedef __attribute__((ext_vector_type(2))) float v2f;
typedef __attribute__((ext_vector_type(8))) float v8f;

// ------------------------------------------------------------------
// GEMM1: feat[4096,256] = tanh(x[4096,64] @ Wpt[64,256] + bpt)
// one wave computes one 16x16 tile via v_wmma_f32_16x16x4_f32
// ------------------------------------------------------------------
__global__ void k_gemm1(const float* __restrict__ X, const float* __restrict__ W,
                        const float* __restrict__ bias, float* __restrict__ feat) {
    int wave = (blockIdx.x * blockDim.x + threadIdx.x) >> 5;
    int lane = threadIdx.x & 31;
    const int tiles_n = D_HID / 16;                 // 16
    int row0 = (wave / tiles_n) * 16;
    int col0 = (wave % tiles_n) * 16;
    int half = lane >> 4;
    int l    = lane & 15;

    v8f c = {};
    for (int k = 0; k < D_IN; k += 4) {
        int ka = k + 2 * half;
        v2f a, b;
        a.x = X[(row0 + l) * D_IN + ka];
        a.y = X[(row0 + l) * D_IN + ka + 1];
        b.x = W[ka * D_HID + col0 + l];
        b.y = W[(ka + 1) * D_HID + col0 + l];
        c = __builtin_amdgcn_wmma_f32_16x16x4_f32(false, a, false, b,
                                                  (short)0, c, false, false);
    }
    float bv = bias[col0 + l];
#pragma unroll
    for (int r = 0; r < 8; ++r) {
        int m = r + 8 * half;
        feat[(row0 + m) * D_HID + col0 + l] = tanhf(c[r] + bv);
    }
}

// ------------------------------------------------------------------
// GEMM2: net[4096,217] = feat[4096,256] @ Wl[256,217] + bl
// N padded to 224 (14 tiles); guard col < 217
// ------------------------------------------------------------------
__global__ void k_gemm2(const float* __restrict__ F, const float* __restrict__ W,
                        const float* __restrict__ bias, float* __restrict__ net) {
    int wave = (blockIdx.x * blockDim.x + threadIdx.x) >> 5;
    int lane = threadIdx.x & 31;
    const int tiles_n = 14;
    int row0 = (wave / tiles_n) * 16;
    int col0 = (wave % tiles_n) * 16;
    int half = lane >> 4;
    int l    = lane & 15;
    int col  = col0 + l;
    bool cok = (col < OUTC);

    v8f c = {};
    for (int k = 0; k < D_HID; k += 4) {
        int ka = k + 2 * half;
        v2f a, b;
        a.x = F[(row0 + l) * D_HID + ka];
        a.y = F[(row0 + l) * D_HID + ka + 1];
        b.x = cok ? W[ka * OUTC + col] : 0.0f;
        b.y = cok ? W[(ka + 1) * OUTC + col] : 0.0f;
        c = __builtin_amdgcn_wmma_f32_16x16x4_f32(false, a, false, b,
                                                  (short)0, c, false, false);
    }
    if (cok) {
        float bv = bias[col];
#pragma unroll
        for (int r = 0; r < 8; ++r) {
            int m = r + 8 * half;
            net[(row0 + m) * OUTC + col] = c[r] + bv;
        }
    }
}

// ------------------------------------------------------------------
// Impulse-response kernel: 1 wave, 32 lanes.
// Lanes 0..29: impulse response of basis n through the 2x2 LTI system.
// Lane 30: homogeneous response of u0; lane 31: of z0.
// G stored as B-matrix [32 rows (basis, padded)][64 cols (checkpoint, padded)]
// ------------------------------------------------------------------
__global__ void k_setup(const float* __restrict__ cArr, const float* __restrict__ hArr,
                        float* __restrict__ G, float* __restrict__ A00,
                        float* __restrict__ A01) {
    int lane = threadIdx.x;
    for (int j = 0; j < 64; ++j) G[lane * 64 + j] = 0.0f;
    if (lane == 0) for (int j = 0; j < 64; ++j) A00[j] = 0.0f;
    if (lane == 1) for (int j = 0; j < 64; ++j) A01[j] = 0.0f;
    if (lane == 0) A00[0] = 1.0f;   // checkpoint 0: y = y0 exactly

    const float cc = (lane < NB) ? cArr[lane] : 0.0f;
    const float hh = (lane < NB) ? hArr[lane] : 0.0f;
    const float d  = DT / TAU;
    const float bz = A_Z / 4.0f;

    float u = (lane == 30) ? 1.0f : 0.0f;
    float z = (lane == 31) ? 1.0f : 0.0f;
    float x = 1.0f;

    for (int t = 1; t <= STEPS; ++t) {
        x = x + (-A_X * x / TAU) * DT;                 // canonical system
        float dx  = x - cc;
        float psi = (lane < NB) ? expf(-hh * dx * dx) : 0.0f;
        float s = psi;                                  // wave32 reduction
        for (int off = 16; off; off >>= 1) s += __shfl_xor(s, off, 32);
        float p  = psi * x / s;                         // forcing basis value
        float un = u + d * z;
        float zn = z + d * (-(A_Z * bz) * u - A_Z * z + p);
        u = un; z = zn;
        if (t % 10 == 0) {
            int chk = t / 10;
            if (lane < NB)       G[lane * 64 + chk] = u;
            else if (lane == 30) A00[chk] = u;
            else                 A01[chk] = u;
        }
    }
}

// ------------------------------------------------------------------
// Combine: Q[e,chk] = w[e,:] . G[:,chk] via WMMA, then
// out[b,chk,dof] = goal + A00*(y0-goal) + A01*0.05 + (goal-y0)*Q
// w[e][n] = net[(e/7)*217 + 7 + (e%7)*30 + n]
// ------------------------------------------------------------------
__global__ void k_combine(const float* __restrict__ net, const float* __restrict__ state,
                          const float* __restrict__ G, const float* __restrict__ A00,
                          const float* __restrict__ A01, float* __restrict__ out) {
    int wave = (blockIdx.x * blockDim.x + threadIdx.x) >> 5;
    int lane = threadIdx.x & 31;
    const int tiles_n = 4;                          // 64 cols cover 51 checkpoints
    int row0 = (wave / tiles_n) * 16;
    int col0 = (wave % tiles_n) * 16;
    int half = lane >> 4;
    int l    = lane & 15;

    // A-fragment row pointer for this lane (row = element index)
    int eA = row0 + l;
    const float* wrow = net + (eA / DOF) * OUTC + DOF + (eA % DOF) * NB;

    v8f c = {};
    for (int k = 0; k < 32; k += 4) {
        int ka = k + 2 * half;
        v2f a, b;
        a.x = (ka     < NB) ? wrow[ka]     : 0.0f;
        a.y = (ka + 1 < NB) ? wrow[ka + 1] : 0.0f;
        b.x = G[ka * 64 + col0 + l];                // rows 30,31 are zero-padded
        b.y = G[(ka + 1) * 64 + col0 + l];
        c = __builtin_amdgcn_wmma_f32_16x16x4_f32(false, a, false, b,
                                                  (short)0, c, false, false);
    }

    int chk = col0 + l;
    if (chk < NCHK) {
        float a00 = A00[chk];
        float a01 = A01[chk];
#pragma unroll
        for (int r = 0; r < 8; ++r) {
            int m = r + 8 * half;
            int e = row0 + m;
            int b_  = e / DOF;
            int dof = e % DOF;
            float goal = net[b_ * OUTC + dof];
            float y0   = state[e];
            float val  = goal + a00 * (y0 - goal) + a01 * 0.05f
                       + (goal - y0) * c[r];
            out[b_ * (NCHK * DOF) + chk * DOF + dof] = val;
        }
    }
}

// ------------------------------------------------------------------
extern "C" void kernel_launch(void* const* d_in, const int* in_sizes, int n_in,
                              void* d_out, int out_size, void* d_ws, size_t ws_size,
                              hipStream_t stream) {
    const float* x     = (const float*)d_in[0];
    const float* state = (const float*)d_in[1];
    const float* Wpt   = (const float*)d_in[2];
    const float* bpt   = (const float*)d_in[3];
    const float* Wl    = (const float*)d_in[4];
    const float* bl    = (const float*)d_in[5];
    const float* cbuf  = (const float*)d_in[6];
    const float* hbuf  = (const float*)d_in[7];
    float* out = (float*)d_out;

    float* feat = (float*)d_ws;                   // 4096*256
    float* net  = feat + BATCH * D_HID;           // 4096*217
    float* G    = net  + BATCH * OUTC;            // 32*64
    float* A00  = G    + 32 * 64;                 // 64
    float* A01  = A00  + 64;                      // 64

    // GEMM1: 256x16 tiles = 4096 waves, 4 waves/block
    k_gemm1<<<1024, 128, 0, stream>>>(x, Wpt, bpt, feat);
    // impulse responses (independent of GEMMs; stream order keeps it before combine)
    k_setup<<<1, 32, 0, stream>>>(cbuf, hbuf, G, A00, A01);
    // GEMM2: 256x14 tiles = 3584 waves
    k_gemm2<<<896, 128, 0, stream>>>(feat, Wl, bl, net);
    // combine: 1792x4 tiles = 7168 waves
    k_combine<<<1792, 128, 0, stream>>>(net, state, G, A00, A01, out);
}